// GCEALsHead_89283780149361
// MI455X (gfx1250) — compile-verified
//
#include <hip/hip_runtime.h>
#include <math.h>

// Mahalanobis soft-assignment: dist = [z^2, z] @ [w ; -2*mu*w] + c[k], then row softmax.
// B=8192, K=256, D=128. Fused reduction length R = 2*D = 256.

#define BATCH 8192
#define KC    256      // clusters
#define DD    128      // feature dim
#define RT    256      // fused reduction = 2*DD
#define ROWS  32       // z-rows per workgroup
#define AP    260      // padded LDS stride for fused [z^2|z] tile (bank-clean b64 loads)
#define OPAD  260      // padded LDS stride for dist tile

typedef __attribute__((ext_vector_type(2))) float v2f;
typedef __attribute__((ext_vector_type(8))) float v8f;

// ---------------------------------------------------------------------------
// Prep: build swizzled B' operand (RT x KC) in WMMA-B fragment order, plus c[k].
// B'[r][n] = (r <  DD) ?  exp(-logvar[n][r])
//          : (r >= DD) ? -2 * centroids[n][r-DD] * exp(-logvar[n][r-DD])
// Fragment order: for r-quad q, half h, col n, the two floats {B'[4q+2h][n],
// B'[4q+2h+1][n]} are stored contiguously -> one global_load_b64 per lane.
// c[n] = sum_d centroids[n][d]^2 * w[n][d]   (deterministic tree reduction)
// ---------------------------------------------------------------------------
__global__ void build_ops_kernel(const float* __restrict__ centroids,
                                 const float* __restrict__ logvar,
                                 float* __restrict__ bswz,
                                 float* __restrict__ cvec) {
    const int n = blockIdx.x;    // cluster 0..255
    const int r = threadIdx.x;   // fused reduction index 0..255
    const int d = r & (DD - 1);
    const float w  = __expf(-logvar[n * DD + d]);
    const float mu = centroids[n * DD + d];
    const float val = (r < DD) ? w : (-2.0f * mu * w);

    const int q = r >> 2, h = (r >> 1) & 1, j = r & 1;
    bswz[(((q * 2 + h) * KC + n) << 1) + j] = val;

    __shared__ float red[DD];
    if (r < DD) red[r] = mu * mu * w;
    __syncthreads();
    for (int s = DD / 2; s > 0; s >>= 1) {
        if (r < s) red[r] += red[r + s];
        __syncthreads();
    }
    if (r == 0) cvec[n] = red[0];
}

// ---------------------------------------------------------------------------
// Main: per-WG 32 rows x 256 clusters. GEMM via v_wmma_f32_16x16x4_f32 over the
// fused reduction (A' = [z^2 | z] staged once in LDS), then in-LDS row softmax
// (the whole K=256 softmax row lives in one workgroup -> single pass).
// ---------------------------------------------------------------------------
__global__ void __launch_bounds__(256)
dist_softmax_kernel(const float* __restrict__ z,
                    const float* __restrict__ bswz,
                    const float* __restrict__ cvec,
                    float* __restrict__ out) {
    __shared__ float la[ROWS * AP];    // fused [z^2 | z] tile, padded
    __shared__ float lo[ROWS * OPAD];  // -0.5*dist tile, padded

    const int t       = threadIdx.x;          // 0..255
    const int wave    = t >> 5;                // 0..7 (wave32)
    const int lane    = t & 31;
    const int h       = lane >> 4;             // lane half
    const int m       = lane & 15;
    const int rowBase = blockIdx.x * ROWS;
    const int colBase = wave * 32;             // 32 K-columns per wave

    // Stage z tile once; write both z^2 (cols 0..127) and z (cols 128..255).
    for (int i = t; i < ROWS * DD; i += 256) {
        const int rr = i >> 7, cc = i & (DD - 1);
        const float v = z[(rowBase + rr) * DD + cc];
        la[rr * AP + cc]      = v * v;
        la[rr * AP + DD + cc] = v;
    }
    __syncthreads();

    v8f acc00 = {}, acc01 = {}, acc10 = {}, acc11 = {};
    const v2f* __restrict__ bp = (const v2f*)bswz;

    // Pure load+WMMA inner loop over the fused reduction (no selects, no VALU math).
#pragma unroll 4
    for (int q = 0; q < RT / 4; ++q) {
        const int dbase = (q << 2) + 2 * h;
        // A fragments: lane m holds {A'[m][dbase], A'[m][dbase+1]} -> ds_load_b64
        v2f a0 = *(const v2f*)&la[(0  + m) * AP + dbase];
        v2f a1 = *(const v2f*)&la[(16 + m) * AP + dbase];
        // B fragments: single coalesced global_load_b64 each (L2-resident)
        const int bq = (q * 2 + h) * KC;
        v2f b0 = bp[bq + colBase + m];
        v2f b1 = bp[bq + colBase + 16 + m];

        acc00 = __builtin_amdgcn_wmma_f32_16x16x4_f32(false, a0, false, b0,
                                                      (short)0, acc00, false, false);
        acc01 = __builtin_amdgcn_wmma_f32_16x16x4_f32(false, a0, false, b1,
                                                      (short)0, acc01, false, false);
        acc10 = __builtin_amdgcn_wmma_f32_16x16x4_f32(false, a1, false, b0,
                                                      (short)0, acc10, false, false);
        acc11 = __builtin_amdgcn_wmma_f32_16x16x4_f32(false, a1, false, b1,
                                                      (short)0, acc11, false, false);
    }

    // Fold in c[k], scale by -0.5, scatter C tiles to LDS.
    // C layout: VGPR j holds row (j + 8*h), col (lane%16) of the 16x16 tile.
    const float c0 = cvec[colBase + m];
    const float c1 = cvec[colBase + 16 + m];
#pragma unroll
    for (int j = 0; j < 8; ++j) {
        const int row0 = h * 8 + j;
        lo[(row0)      * OPAD + colBase + m]      = -0.5f * (acc00[j] + c0);
        lo[(row0)      * OPAD + colBase + 16 + m] = -0.5f * (acc01[j] + c1);
        lo[(row0 + 16) * OPAD + colBase + m]      = -0.5f * (acc10[j] + c0);
        lo[(row0 + 16) * OPAD + colBase + 16 + m] = -0.5f * (acc11[j] + c1);
    }
    __syncthreads();

    // Stable softmax over K=256: wave w owns local rows 4w..4w+3.
#pragma unroll
    for (int rr = 0; rr < 4; ++rr) {
        const int row = wave * 4 + rr;
        float v[8];
        float mx = -INFINITY;
#pragma unroll
        for (int i = 0; i < 8; ++i) {
            v[i] = lo[row * OPAD + lane + 32 * i];
            mx = fmaxf(mx, v[i]);
        }
        for (int off = 16; off > 0; off >>= 1)
            mx = fmaxf(mx, __shfl_xor(mx, off, 32));
        float s = 0.0f;
#pragma unroll
        for (int i = 0; i < 8; ++i) {
            v[i] = __expf(v[i] - mx);
            s += v[i];
        }
        for (int off = 16; off > 0; off >>= 1)
            s += __shfl_xor(s, off, 32);
        const float inv = 1.0f / s;
#pragma unroll
        for (int i = 0; i < 8; ++i)
            out[(rowBase + row) * KC + lane + 32 * i] = v[i] * inv;
    }
}

// ---------------------------------------------------------------------------
extern "C" void kernel_launch(void* const* d_in, const int* in_sizes, int n_in,
                              void* d_out, int out_size, void* d_ws, size_t ws_size,
                              hipStream_t stream) {
    (void)in_sizes; (void)n_in; (void)out_size; (void)ws_size;
    const float* z         = (const float*)d_in[0];   // (8192, 128)
    const float* centroids = (const float*)d_in[1];   // (256, 128)
    const float* logvar    = (const float*)d_in[2];   // (256, 128)
    float* out = (float*)d_out;                       // (8192, 256)

    float* bswz = (float*)d_ws;                       // RT*KC floats = 256 KB
    float* cvec = bswz + RT * KC;                     // 256 floats

    build_ops_kernel<<<dim3(KC), dim3(RT), 0, stream>>>(centroids, logvar, bswz, cvec);
    dist_softmax_kernel<<<dim3(BATCH / ROWS), dim3(256), 0, stream>>>(z, bswz, cvec, out);
}